// LNLieBracketChannelMix_62019327754501
// MI455X (gfx1250) — compile-verified
//
#include <hip/hip_runtime.h>

// CDNA5 / gfx1250, wave32.
typedef __attribute__((ext_vector_type(2))) float v2f;
typedef __attribute__((ext_vector_type(8))) float v8f;

#define F_DIM 256
#define K_DIM 3
#define N_DIM 4096

// Workgroup: 256 threads = 8 waves, arranged 4 (g-tiles) x 2 (n-tiles).
// Each wave: one 16x16 (g,n) tile, 6 f32 WMMA accumulators (3 k x 2 weights).
__global__ __launch_bounds__(256)
void lie_bracket_channel_mix_kernel(const float* __restrict__ x,
                                    const float* __restrict__ M1,
                                    const float* __restrict__ M2,
                                    const float* __restrict__ W1,
                                    const float* __restrict__ W2,
                                    float* __restrict__ out)
{
    const int lane = threadIdx.x & 31;   // wave32
    const int wave = threadIdx.x >> 5;   // 0..7
    const int half = lane >> 4;          // 0 or 1
    const int lm   = lane & 15;

    const int b     = blockIdx.z;
    const int gwave = blockIdx.y * 64 + (wave >> 1) * 16;  // output-channel tile base
    const int nwave = blockIdx.x * 32 + (wave & 1) * 16;   // n tile base

    // A-matrix (16x4 f32) per-lane layout: lanes 0-15 hold K=0,1; lanes 16-31 hold K=2,3.
    const int kbase = half * 2;

    const float* a1ptr = W1 + (size_t)(gwave + lm) * F_DIM + kbase;
    const float* a2ptr = W2 + (size_t)(gwave + lm) * F_DIM + kbase;

    // x[((b*F + f)*K + k)*N + n]; per-lane B-fragment base (column nwave+lm).
    const float* xb = x + (size_t)b * F_DIM * K_DIM * N_DIM + (size_t)nwave + lm;

    v8f acc[2][3];
#pragma unroll
    for (int w = 0; w < 2; ++w)
#pragma unroll
        for (int k = 0; k < 3; ++k)
            acc[w][k] = (v8f)(0.0f);

    // GEMM over the channel dimension f (the WMMA K-dim), 4 per step.
#pragma unroll 4
    for (int f = 0; f < F_DIM; f += 4) {
        // A fragments: W[(g)*F + f + kbase .. +1]  (8B-aligned b64 load)
        v2f a1 = *(const v2f*)(a1ptr + f);
        v2f a2 = *(const v2f*)(a2ptr + f);

        // B fragments (4x16 f32), mirroring A layout: lane half selects K pair.
        v2f bf[3];
#pragma unroll
        for (int k = 0; k < 3; ++k) {
            const float* p = xb + ((size_t)(f + kbase) * K_DIM + k) * N_DIM;
            bf[k].x = p[0];
            bf[k].y = p[(size_t)K_DIM * N_DIM];   // next f row
        }

#pragma unroll
        for (int k = 0; k < 3; ++k) {
            acc[0][k] = __builtin_amdgcn_wmma_f32_16x16x4_f32(
                false, a1, false, bf[k], (short)0, acc[0][k], false, false);
            acc[1][k] = __builtin_amdgcn_wmma_f32_16x16x4_f32(
                false, a2, false, bf[k], (short)0, acc[1][k], false, false);
        }
    }

    // 3x3 rotation matrices (uniform -> scalar loads).
    float m1[9], m2[9];
#pragma unroll
    for (int i = 0; i < 9; ++i) { m1[i] = M1[i]; m2[i] = M2[i]; }

    // Epilogue: per C/D layout, VGPR j holds row M = j (+8 for lanes 16-31), col = lm.
#pragma unroll
    for (int j = 0; j < 8; ++j) {
        const int g = gwave + j + 8 * half;
        const int n = nwave + lm;

        const float d10 = acc[0][0][j], d11 = acc[0][1][j], d12 = acc[0][2][j];
        const float d20 = acc[1][0][j], d21 = acc[1][1][j], d22 = acc[1][2][j];

        // drot = M1 @ d1 ; erot = M2 @ d2
        const float dr0 = m1[0]*d10 + m1[1]*d11 + m1[2]*d12;
        const float dr1 = m1[3]*d10 + m1[4]*d11 + m1[5]*d12;
        const float dr2 = m1[6]*d10 + m1[7]*d11 + m1[8]*d12;
        const float er0 = m2[0]*d20 + m2[1]*d21 + m2[2]*d22;
        const float er1 = m2[3]*d20 + m2[4]*d21 + m2[5]*d22;
        const float er2 = m2[6]*d20 + m2[7]*d21 + m2[8]*d22;

        // vee(e_hat@d_hat - d_hat@e_hat) = cross(e, d)
        const float v0 = er1 * dr2 - er2 * dr1;
        const float v1 = er2 * dr0 - er0 * dr2;
        const float v2 = er0 * dr1 - er1 * dr0;

        const size_t base = ((size_t)(b * F_DIM + g) * K_DIM) * N_DIM + n;
        out[base]                     = x[base]                     + v0;
        out[base + N_DIM]             = x[base + N_DIM]             + v1;
        out[base + 2 * (size_t)N_DIM] = x[base + 2 * (size_t)N_DIM] + v2;
    }
}

extern "C" void kernel_launch(void* const* d_in, const int* in_sizes, int n_in,
                              void* d_out, int out_size, void* d_ws, size_t ws_size,
                              hipStream_t stream) {
    const float* x  = (const float*)d_in[0];
    const float* M1 = (const float*)d_in[1];
    const float* M2 = (const float*)d_in[2];
    const float* W1 = (const float*)d_in[3];
    const float* W2 = (const float*)d_in[4];
    float* out = (float*)d_out;

    const int B = in_sizes[0] / (F_DIM * K_DIM * N_DIM);  // = 8

    dim3 grid(N_DIM / 32, F_DIM / 64, B);
    dim3 block(256, 1, 1);
    hipLaunchKernelGGL(lie_bracket_channel_mix_kernel, grid, block, 0, stream,
                       x, M1, M2, W1, W2, out);
}